// PointNetEncoder_17386027614573
// MI455X (gfx1250) — compile-verified
//
#include <hip/hip_runtime.h>
#include <hip/hip_bf16.h>
#include <math.h>

// ---------------------------------------------------------------------------
// VN-PointNet encoder for MI455X (gfx1250, wave32).
// Channel-mixing linears run on V_WMMA_F32_16X16X4_F32 (fp32 matrix path).
// BN-over-norms: stats GEMM pass + fused recompute/apply GEMM pass (no P/D
// temporaries -> roughly half the HBM traffic of the naive pipeline).
// K-loops are branch-free: zero-padded LDS weight tiles + clamped B addresses.
// ---------------------------------------------------------------------------

#define EPSV   1e-6f
#define BNEPS  1e-5f
#define KC     128      // K-chunk staged in LDS (16 rows x KC floats)

static const int  Bsz  = 8;
static const int  Npts = 4096;
static const int  KNN  = 20;
static const int  PK   = Npts * KNN; // 81920 positions in the (N,k) stage

typedef float v2f __attribute__((ext_vector_type(2)));
typedef float v8f __attribute__((ext_vector_type(8)));

// D(16x16) = A(16x4) * B(4x16) + C, fp32.
// A lane l: M = l&15, K = (l>>4)*2 + r   (r = vgpr index 0/1)
// B lane l: N = l&15, K = (l>>4)*2 + r
// C/D lane l, vgpr r: M = r + (l>>4)*8, N = l&15
__device__ __forceinline__ v8f wmma_f32_16x16x4(v2f a, v2f b, v8f c) {
#if defined(__HIP_DEVICE_COMPILE__) && __has_builtin(__builtin_amdgcn_wmma_f32_16x16x4_f32)
  return __builtin_amdgcn_wmma_f32_16x16x4_f32(false, a, false, b, (short)0, c,
                                               false, false);
#else
  c[0] += a[0] * b[0]; // placeholder so host pass parses; never used on device
  return c;
#endif
}

// ------------------------------- KNN ---------------------------------------
__global__ __launch_bounds__(256)
void knn_kernel(const float* __restrict__ x, int* __restrict__ idxOut) {
  const int b = blockIdx.y;
  const int i = blockIdx.x * 256 + threadIdx.x;
  const float* xb = x + (size_t)b * 3 * Npts;
  const float xi0 = xb[i], xi1 = xb[Npts + i], xi2 = xb[2 * Npts + i];
  const float xxi = xi0 * xi0 + xi1 * xi1 + xi2 * xi2;
  __shared__ float s0[256], s1[256], s2[256], sq[256];
  float bv[KNN]; int bi[KNN];
#pragma unroll
  for (int k = 0; k < KNN; ++k) { bv[k] = -3.0e38f; bi[k] = 0; }
  float wmin = -3.0e38f; int wpos = 0;
  for (int jt = 0; jt < Npts; jt += 256) {
    __syncthreads();
    const int j = jt + threadIdx.x;
    const float a0 = xb[j], a1 = xb[Npts + j], a2 = xb[2 * Npts + j];
    s0[threadIdx.x] = a0; s1[threadIdx.x] = a1; s2[threadIdx.x] = a2;
    sq[threadIdx.x] = a0 * a0 + a1 * a1 + a2 * a2;
    __syncthreads();
    for (int t = 0; t < 256; ++t) {
      const float pd = 2.f * (xi0 * s0[t] + xi1 * s1[t] + xi2 * s2[t]) - xxi - sq[t];
      if (pd > wmin) {
        bv[wpos] = pd; bi[wpos] = jt + t;
        wmin = bv[0]; wpos = 0;
#pragma unroll
        for (int k = 1; k < KNN; ++k)
          if (bv[k] < wmin) { wmin = bv[k]; wpos = k; }
      }
    }
  }
  int* o = idxOut + ((size_t)b * Npts + i) * KNN;
#pragma unroll
  for (int k = 0; k < KNN; ++k) o[k] = bi[k];
}

// ------------------------- edge features (cross) ----------------------------
__global__ void feat_kernel(const float* __restrict__ x, const int* __restrict__ idx,
                            float* __restrict__ feat) {
  const long t = (long)blockIdx.x * blockDim.x + threadIdx.x;
  if (t >= (long)Bsz * Npts * KNN) return;
  const int kk = (int)(t % KNN);
  const long r = t / KNN;
  const int n = (int)(r % Npts);
  const int b = (int)(r / Npts);
  const float* xb = x + (size_t)b * 3 * Npts;
  const int j = idx[((size_t)b * Npts + n) * KNN + kk];
  const float c0 = xb[n], c1 = xb[Npts + n], c2 = xb[2 * Npts + n];
  const float p0 = xb[j], p1 = xb[Npts + j], p2 = xb[2 * Npts + j];
  const float cr0 = p1 * c2 - p2 * c1;
  const float cr1 = p2 * c0 - p0 * c2;
  const float cr2 = p0 * c1 - p1 * c0;
  float* f = feat + (size_t)b * 9 * PK + (size_t)n * KNN + kk;
  f[0 * (size_t)PK] = p0 - c0; f[1 * (size_t)PK] = p1 - c1; f[2 * (size_t)PK] = p2 - c2;
  f[3 * (size_t)PK] = c0;      f[4 * (size_t)PK] = c1;      f[5 * (size_t)PK] = c2;
  f[6 * (size_t)PK] = cr0;     f[7 * (size_t)PK] = cr1;     f[8 * (size_t)PK] = cr2;
}

// ---------------- WMMA GEMM pass 1: per-channel norm statistics -------------
// X layout per batch: [Cin][3][P]; computes p = Wf*X for v=0,1,2 tiles and
// accumulates sum / sum^2 of (||p||+EPS) per output channel.
__global__ __launch_bounds__(128)
void vn_gemm_stats(const float* __restrict__ X, const float* __restrict__ Wf,
                   float* __restrict__ stats, int O, int Cin, int P,
                   long long xBS) {
  __shared__ float sF[16 * KC];
  const int lane = threadIdx.x & 31;
  const int wave = threadIdx.x >> 5;
  const int half = lane >> 4;
  const int l16  = lane & 15;
  const int o0   = blockIdx.x * 16;
  const int col  = (blockIdx.y * 4 + wave) * 16 + l16;
  const int b    = blockIdx.z;
  const float* Xb = X + (size_t)b * xBS + col;
  v8f a0 = {}, a1 = {}, a2 = {};
  for (int kc0 = 0; kc0 < Cin; kc0 += KC) {
    const int kcLen = min(KC, Cin - kc0);
    for (int idx = threadIdx.x; idx < 16 * KC; idx += 128) {
      const int row = idx >> 7;          // KC == 128
      const int kk  = idx & (KC - 1);
      sF[idx] = (kk < kcLen)
          ? Wf[(size_t)min(o0 + row, O - 1) * Cin + kc0 + kk] : 0.f;
    }
    __syncthreads();
    const int kcA = kcLen & ~3;
    const float* pB = Xb + (size_t)(kc0 + half * 2) * 3 * P;
    const float* pF = &sF[l16 * KC + half * 2];
    for (int k0 = 0; k0 < kcA; k0 += 4) {
      const v2f af = *(const v2f*)pF;
      v2f b0, b1, b2;
      b0[0] = pB[0];                 b0[1] = pB[3 * (size_t)P];
      b1[0] = pB[(size_t)P];         b1[1] = pB[4 * (size_t)P];
      b2[0] = pB[2 * (size_t)P];     b2[1] = pB[5 * (size_t)P];
      a0 = wmma_f32_16x16x4(af, b0, a0);
      a1 = wmma_f32_16x16x4(af, b1, a1);
      a2 = wmma_f32_16x16x4(af, b2, a2);
      pF += 4; pB += 12 * (size_t)P;
    }
    if (kcA < kcLen) {                  // tail: A zero-padded, B clamped
      const v2f af = *(const v2f*)&sF[l16 * KC + kcA + half * 2];
      v2f b0, b1, b2;
#pragma unroll
      for (int r = 0; r < 2; ++r) {
        const size_t xo = (size_t)min(kc0 + kcA + half * 2 + r, Cin - 1) * 3 * P;
        b0[r] = Xb[xo]; b1[r] = Xb[xo + (size_t)P]; b2[r] = Xb[xo + 2 * (size_t)P];
      }
      a0 = wmma_f32_16x16x4(af, b0, a0);
      a1 = wmma_f32_16x16x4(af, b1, a1);
      a2 = wmma_f32_16x16x4(af, b2, a2);
    }
    __syncthreads();
  }
#pragma unroll
  for (int r = 0; r < 8; ++r) {
    const int o = o0 + r + half * 8;
    const float p0 = a0[r], p1 = a1[r], p2 = a2[r];
    const float nrm = sqrtf(p0 * p0 + p1 * p1 + p2 * p2) + EPSV;
    const bool valid = (o < O);
    float s  = valid ? nrm : 0.f;
    float s2 = valid ? nrm * nrm : 0.f;
#pragma unroll
    for (int d = 1; d < 16; d <<= 1) {   // reduce within each 16-lane half
      s  += __shfl_xor(s, d);
      s2 += __shfl_xor(s2, d);
    }
    if (l16 == 0 && valid) {
      atomicAdd(&stats[o], s);
      atomicAdd(&stats[O + o], s2);
    }
  }
}

__global__ void bn_finalize(float* __restrict__ stats, int O, float invCount) {
  const int o = blockIdx.x * blockDim.x + threadIdx.x;
  if (o >= O) return;
  const float mean = stats[o] * invCount;
  const float var  = stats[O + o] * invCount - mean * mean;
  stats[o]     = mean;
  stats[O + o] = 1.f / sqrtf(var + BNEPS);
}

// --------- WMMA GEMM pass 2: recompute p (and d), BN + leaky, store ---------
// MODE: 0 = bn + projection-leaky-relu (needs Wd), 1 = bn only, 2 = plain lin
template <int MODE>
__global__ __launch_bounds__(128)
void vn_gemm_apply(const float* __restrict__ X, const float* __restrict__ Wf,
                   const float* __restrict__ Wd, const float* __restrict__ stats,
                   float* __restrict__ out, int O, int Cin, int P,
                   long long xBS, long long outBS) {
  __shared__ float sF[16 * KC];
  __shared__ float sD[(MODE == 0) ? 16 * KC : 1];
  const int lane = threadIdx.x & 31;
  const int wave = threadIdx.x >> 5;
  const int half = lane >> 4;
  const int l16  = lane & 15;
  const int o0   = blockIdx.x * 16;
  const int col  = (blockIdx.y * 4 + wave) * 16 + l16;
  const int b    = blockIdx.z;
  const float* Xb = X + (size_t)b * xBS + col;
  v8f p0a = {}, p1a = {}, p2a = {}, d0a = {}, d1a = {}, d2a = {};
  for (int kc0 = 0; kc0 < Cin; kc0 += KC) {
    const int kcLen = min(KC, Cin - kc0);
    for (int idx = threadIdx.x; idx < 16 * KC; idx += 128) {
      const int row = idx >> 7;
      const int kk  = idx & (KC - 1);
      const size_t g = (size_t)min(o0 + row, O - 1) * Cin + kc0 + kk;
      const bool kv = kk < kcLen;
      sF[idx] = kv ? Wf[g] : 0.f;
      if constexpr (MODE == 0) sD[idx] = kv ? Wd[g] : 0.f;
    }
    __syncthreads();
    const int kcA = kcLen & ~3;
    const float* pB = Xb + (size_t)(kc0 + half * 2) * 3 * P;
    const float* pF = &sF[l16 * KC + half * 2];
    const float* pD = &sD[(MODE == 0) ? (l16 * KC + half * 2) : 0];
    for (int k0 = 0; k0 < kcA; k0 += 4) {
      const v2f af = *(const v2f*)pF;
      v2f b0, b1, b2;
      b0[0] = pB[0];                 b0[1] = pB[3 * (size_t)P];
      b1[0] = pB[(size_t)P];         b1[1] = pB[4 * (size_t)P];
      b2[0] = pB[2 * (size_t)P];     b2[1] = pB[5 * (size_t)P];
      p0a = wmma_f32_16x16x4(af, b0, p0a);
      p1a = wmma_f32_16x16x4(af, b1, p1a);
      p2a = wmma_f32_16x16x4(af, b2, p2a);
      if constexpr (MODE == 0) {
        const v2f ad = *(const v2f*)pD;
        d0a = wmma_f32_16x16x4(ad, b0, d0a);
        d1a = wmma_f32_16x16x4(ad, b1, d1a);
        d2a = wmma_f32_16x16x4(ad, b2, d2a);
        pD += 4;
      }
      pF += 4; pB += 12 * (size_t)P;
    }
    if (kcA < kcLen) {                  // tail: A zero-padded, B clamped
      const v2f af = *(const v2f*)&sF[l16 * KC + kcA + half * 2];
      v2f b0, b1, b2;
#pragma unroll
      for (int r = 0; r < 2; ++r) {
        const size_t xo = (size_t)min(kc0 + kcA + half * 2 + r, Cin - 1) * 3 * P;
        b0[r] = Xb[xo]; b1[r] = Xb[xo + (size_t)P]; b2[r] = Xb[xo + 2 * (size_t)P];
      }
      p0a = wmma_f32_16x16x4(af, b0, p0a);
      p1a = wmma_f32_16x16x4(af, b1, p1a);
      p2a = wmma_f32_16x16x4(af, b2, p2a);
      if constexpr (MODE == 0) {
        const v2f ad = *(const v2f*)&sD[l16 * KC + kcA + half * 2];
        d0a = wmma_f32_16x16x4(ad, b0, d0a);
        d1a = wmma_f32_16x16x4(ad, b1, d1a);
        d2a = wmma_f32_16x16x4(ad, b2, d2a);
      }
    }
    __syncthreads();
  }
#pragma unroll
  for (int r = 0; r < 8; ++r) {
    const int o = o0 + r + half * 8;
    if (o >= O) continue;
    float q0 = p0a[r], q1 = p1a[r], q2 = p2a[r];
    if constexpr (MODE != 2) {
      const float nrm   = sqrtf(q0 * q0 + q1 * q1 + q2 * q2) + EPSV;
      const float scale = ((nrm - stats[o]) * stats[O + o]) / nrm;
      q0 *= scale; q1 *= scale; q2 *= scale;
      if constexpr (MODE == 0) {
        const float dd0 = d0a[r], dd1 = d1a[r], dd2 = d2a[r];
        const float dot = q0 * dd0 + q1 * dd1 + q2 * dd2;
        if (dot < 0.f) {
          const float f = dot / (dd0 * dd0 + dd1 * dd1 + dd2 * dd2 + EPSV);
          q0 -= f * dd0; q1 -= f * dd1; q2 -= f * dd2;
        }
      }
    }
    float* ob = out + (size_t)b * outBS + (size_t)o * 3 * P + col;
    ob[0] = q0; ob[(size_t)P] = q1; ob[2 * (size_t)P] = q2;
  }
}

// ---------------------- max-pool over k (pos stage) -------------------------
__global__ __launch_bounds__(256)
void pool_k_kernel(const float* __restrict__ h1, const float* __restrict__ Wd,
                   float* __restrict__ out) {
  const int n = blockIdx.x, b = blockIdx.y;
  __shared__ float xl[21 * 60];   // [o][v][kk]
  __shared__ float dots[21 * 20];
  const float* hb = h1 + (size_t)b * 21 * 3 * PK + (size_t)n * KNN;
  for (int idx = threadIdx.x; idx < 1260; idx += 256) {
    const int o = idx / 60, rem = idx % 60, v = rem / 20, kk = rem % 20;
    xl[idx] = hb[(size_t)(o * 3 + v) * PK + kk];
  }
  __syncthreads();
  for (int idx = threadIdx.x; idx < 420; idx += 256) {
    const int o = idx / 20, kk = idx % 20;
    float d0 = 0, d1 = 0, d2 = 0;
    for (int c = 0; c < 21; ++c) {
      const float w = Wd[o * 21 + c];
      d0 += w * xl[c * 60 + kk];
      d1 += w * xl[c * 60 + 20 + kk];
      d2 += w * xl[c * 60 + 40 + kk];
    }
    dots[idx] = xl[o * 60 + kk] * d0 + xl[o * 60 + 20 + kk] * d1 +
                xl[o * 60 + 40 + kk] * d2;
  }
  __syncthreads();
  if (threadIdx.x < 21) {
    const int o = threadIdx.x;
    float best = dots[o * 20]; int bk = 0;
    for (int kk = 1; kk < 20; ++kk)
      if (dots[o * 20 + kk] > best) { best = dots[o * 20 + kk]; bk = kk; }
    float* ob = out + (size_t)b * 21 * 3 * Npts + (size_t)o * 3 * Npts + n;
    ob[0]        = xl[o * 60 + bk];
    ob[Npts]     = xl[o * 60 + 20 + bk];
    ob[2 * Npts] = xl[o * 60 + 40 + bk];
  }
}

// ---------------------- max-pool over N (st_pool) ---------------------------
__global__ __launch_bounds__(256)
void pool_n_kernel(const float* __restrict__ x, const float* __restrict__ d,
                   float* __restrict__ sv, int C) {
  const int o = blockIdx.x, b = blockIdx.y;
  const float* xb = x + ((size_t)b * C + o) * 3 * Npts;
  const float* db = d + ((size_t)b * C + o) * 3 * Npts;
  float best = -3.0e38f; int bn = 0;
  for (int n = threadIdx.x; n < Npts; n += 256) {
    const float dot = xb[n] * db[n] + xb[Npts + n] * db[Npts + n] +
                      xb[2 * Npts + n] * db[2 * Npts + n];
    if (dot > best) { best = dot; bn = n; }
  }
  __shared__ float rb[256]; __shared__ int ri[256];
  rb[threadIdx.x] = best; ri[threadIdx.x] = bn;
  __syncthreads();
  for (int s = 128; s > 0; s >>= 1) {
    if (threadIdx.x < s && rb[threadIdx.x + s] > rb[threadIdx.x]) {
      rb[threadIdx.x] = rb[threadIdx.x + s]; ri[threadIdx.x] = ri[threadIdx.x + s];
    }
    __syncthreads();
  }
  if (threadIdx.x == 0) {
    const int nb = ri[0];
    float* ob = sv + ((size_t)b * C + o) * 3;
    ob[0] = xb[nb]; ob[1] = xb[Npts + nb]; ob[2] = xb[2 * Npts + nb];
  }
}

// -------------- small (B,C,3) branch: one-block fused VN layer --------------
__global__ __launch_bounds__(512)
void vn_small_kernel(const float* __restrict__ S, const float* __restrict__ Wf,
                     const float* __restrict__ Wd, float* __restrict__ outS,
                     int O, int Cin, int mode) {
  __shared__ float pp[8 * 170 * 3];
  __shared__ float nr[8 * 170];
  __shared__ float mb[170], rbv[170];
  const int BD = blockDim.x;
  for (int idx = threadIdx.x; idx < Bsz * O; idx += BD) {
    const int b = idx / O, o = idx % O;
    const float* sb = S + (size_t)b * Cin * 3;
    const float* w  = Wf + (size_t)o * Cin;
    float p0 = 0, p1 = 0, p2 = 0;
    for (int i = 0; i < Cin; ++i) {
      const float wv = w[i];
      p0 += wv * sb[i * 3]; p1 += wv * sb[i * 3 + 1]; p2 += wv * sb[i * 3 + 2];
    }
    pp[idx * 3] = p0; pp[idx * 3 + 1] = p1; pp[idx * 3 + 2] = p2;
    nr[idx] = sqrtf(p0 * p0 + p1 * p1 + p2 * p2) + EPSV;
  }
  __syncthreads();
  if (mode != 2) {
    for (int o = threadIdx.x; o < O; o += BD) {
      float s = 0, s2 = 0;
      for (int b = 0; b < Bsz; ++b) { const float v = nr[b * O + o]; s += v; s2 += v * v; }
      const float mean = s / (float)Bsz;
      const float var  = s2 / (float)Bsz - mean * mean;
      mb[o] = mean; rbv[o] = 1.f / sqrtf(var + BNEPS);
    }
  }
  __syncthreads();
  for (int idx = threadIdx.x; idx < Bsz * O; idx += BD) {
    const int b = idx / O, o = idx % O;
    float q0 = pp[idx * 3], q1 = pp[idx * 3 + 1], q2 = pp[idx * 3 + 2];
    if (mode != 2) {
      const float scale = ((nr[idx] - mb[o]) * rbv[o]) / nr[idx];
      q0 *= scale; q1 *= scale; q2 *= scale;
      if (mode == 0) {
        const float* sb = S + (size_t)b * Cin * 3;
        const float* w  = Wd + (size_t)o * Cin;
        float d0 = 0, d1 = 0, d2 = 0;
        for (int i = 0; i < Cin; ++i) {
          const float wv = w[i];
          d0 += wv * sb[i * 3]; d1 += wv * sb[i * 3 + 1]; d2 += wv * sb[i * 3 + 2];
        }
        const float dot = q0 * d0 + q1 * d1 + q2 * d2;
        if (dot < 0.f) {
          const float f = dot / (d0 * d0 + d1 * d1 + d2 * d2 + EPSV);
          q0 -= f * d0; q1 -= f * d1; q2 -= f * d2;
        }
      }
    }
    float* ob = outS + (size_t)(b * O + o) * 3;
    ob[0] = q0; ob[1] = q1; ob[2] = q2;
  }
}

// ------------------------------ glue kernels --------------------------------
__global__ void concat42_kernel(const float* __restrict__ h21,
                                const float* __restrict__ s21,
                                float* __restrict__ out) {
  const long t = (long)blockIdx.x * blockDim.x + threadIdx.x;
  if (t >= (long)Bsz * 42 * 3 * Npts) return;
  const int n = (int)(t % Npts);
  long r = t / Npts;
  const int v = (int)(r % 3); r /= 3;
  const int c = (int)(r % 42);
  const int b = (int)(r / 42);
  out[t] = (c < 21)
      ? h21[((size_t)(b * 21 + c) * 3 + v) * Npts + n]
      : s21[(size_t)(b * 21 + (c - 21)) * 3 + v];
}

__global__ __launch_bounds__(256)
void mean_bcast_kernel(float* __restrict__ h682) {
  const int c = blockIdx.x;  // 0..340
  const int b = blockIdx.y;
  float* hb = h682 + (size_t)b * 682 * 3 * Npts;
  __shared__ float red[256];
  for (int v = 0; v < 3; ++v) {
    const float* src = hb + (size_t)(c * 3 + v) * Npts;
    float s = 0;
    for (int n = threadIdx.x; n < Npts; n += 256) s += src[n];
    red[threadIdx.x] = s;
    __syncthreads();
    for (int st = 128; st > 0; st >>= 1) {
      if (threadIdx.x < st) red[threadIdx.x] += red[threadIdx.x + st];
      __syncthreads();
    }
    const float mean = red[0] / (float)Npts;
    float* dst = hb + (size_t)((341 + c) * 3 + v) * Npts;
    for (int n = threadIdx.x; n < Npts; n += 256) dst[n] = mean;
    __syncthreads();
  }
}

__global__ __launch_bounds__(256)
void hstd_max_kernel(const float* __restrict__ h682, const float* __restrict__ z0,
                     float* __restrict__ out) {
  const int i = blockIdx.x;   // 0..681
  const int b = blockIdx.y;
  const float* hb = h682 + (size_t)b * 682 * 3 * Npts + (size_t)i * 3 * Npts;
  const float* zb = z0 + (size_t)b * 9 * Npts;
  float m0 = -3e38f, m1 = -3e38f, m2 = -3e38f;
  for (int n = threadIdx.x; n < Npts; n += 256) {
    const float h0 = hb[n], h1 = hb[Npts + n], h2 = hb[2 * Npts + n];
    m0 = fmaxf(m0, h0 * zb[n]            + h1 * zb[3 * Npts + n] + h2 * zb[6 * Npts + n]);
    m1 = fmaxf(m1, h0 * zb[Npts + n]     + h1 * zb[4 * Npts + n] + h2 * zb[7 * Npts + n]);
    m2 = fmaxf(m2, h0 * zb[2 * Npts + n] + h1 * zb[5 * Npts + n] + h2 * zb[8 * Npts + n]);
  }
  __shared__ float r0[256], r1[256], r2[256];
  r0[threadIdx.x] = m0; r1[threadIdx.x] = m1; r2[threadIdx.x] = m2;
  __syncthreads();
  for (int s = 128; s > 0; s >>= 1) {
    if (threadIdx.x < s) {
      r0[threadIdx.x] = fmaxf(r0[threadIdx.x], r0[threadIdx.x + s]);
      r1[threadIdx.x] = fmaxf(r1[threadIdx.x], r1[threadIdx.x + s]);
      r2[threadIdx.x] = fmaxf(r2[threadIdx.x], r2[threadIdx.x + s]);
    }
    __syncthreads();
  }
  if (threadIdx.x == 0) {
    float* ob = out + (size_t)b * 2046 + (size_t)i * 3;
    ob[0] = r0[0]; ob[1] = r1[0]; ob[2] = r2[0];
  }
}

__global__ void copy_kernel(const float* __restrict__ src, float* __restrict__ dst,
                            long n) {
  const long t = (long)blockIdx.x * blockDim.x + threadIdx.x;
  if (t < n) dst[t] = src[t];
}

// ----------------------------------------------------------------------------
extern "C" void kernel_launch(void* const* d_in, const int* in_sizes, int n_in,
                              void* d_out, int out_size, void* d_ws, size_t ws_size,
                              hipStream_t stream) {
  (void)in_sizes; (void)n_in; (void)out_size; (void)ws_size;
  const float* x        = (const float*)d_in[0];
  const float* pos_Wf   = (const float*)d_in[1];
  const float* pos_Wd   = (const float*)d_in[2];
  const float* pool_Wd  = (const float*)d_in[3];
  const float* c1_Wf    = (const float*)d_in[4];
  const float* c1_Wd    = (const float*)d_in[5];
  const float* st1_Wf   = (const float*)d_in[6];
  const float* st1_Wd   = (const float*)d_in[7];
  const float* st2_Wf   = (const float*)d_in[8];
  const float* st2_Wd   = (const float*)d_in[9];
  const float* st3_Wf   = (const float*)d_in[10];
  const float* st3_Wd   = (const float*)d_in[11];
  const float* stp_Wd   = (const float*)d_in[12];
  const float* stf1_Wf  = (const float*)d_in[13];
  const float* stf1_Wd  = (const float*)d_in[14];
  const float* stf2_Wf  = (const float*)d_in[15];
  const float* stf2_Wd  = (const float*)d_in[16];
  const float* stf3_W   = (const float*)d_in[17];
  const float* c2_Wf    = (const float*)d_in[18];
  const float* c2_Wd    = (const float*)d_in[19];
  const float* c3_W     = (const float*)d_in[20];
  const float* std1_Wf  = (const float*)d_in[21];
  const float* std1_Wd  = (const float*)d_in[22];
  const float* std2_Wf  = (const float*)d_in[23];
  const float* std2_Wd  = (const float*)d_in[24];
  const float* std_lin  = (const float*)d_in[25];

  float* ws = (float*)d_ws;
  size_t off = 0;
  auto alloc = [&](size_t n) { float* p = ws + off; off += n; return p; };
  float* R0   = alloc((size_t)Bsz * 682 * 3 * Npts);  // pos h1 early; h682 late
  float* R1   = alloc((size_t)Bsz * 341 * 3 * Npts);  // s341 (st3 out)
  float* R2   = alloc((size_t)Bsz * 341 * 3 * Npts);  // st_pool d; later z341
  float* R3   = alloc((size_t)Bsz * 170 * 3 * Npts);  // feat early; z170 late
  int*   IDX  = (int*)alloc((size_t)Bsz * Npts * KNN);
  float* h21  = alloc((size_t)Bsz * 21 * 3 * Npts);
  float* h21b = alloc((size_t)Bsz * 21 * 3 * Npts);
  float* sA   = alloc((size_t)Bsz * 21 * 3 * Npts);
  float* sB   = alloc((size_t)Bsz * 42 * 3 * Npts);
  float* hcat = alloc((size_t)Bsz * 42 * 3 * Npts);
  float* h42  = alloc((size_t)Bsz * 42 * 3 * Npts);
  float* sv341 = alloc((size_t)Bsz * 341 * 3);
  float* sv170 = alloc((size_t)Bsz * 170 * 3);
  float* sv85  = alloc((size_t)Bsz * 85 * 3);
  float* sv21  = alloc((size_t)Bsz * 21 * 3);
  float* z0b   = alloc((size_t)Bsz * 9 * Npts);
  float* stats = alloc(2048);

  auto vn_layer = [&](const float* X, const float* Wf, const float* Wd, float* out,
                      int O, int Cin, int P, long long xBS, long long outBS,
                      int mode) {
    dim3 grid((O + 15) / 16, P / 64, Bsz);
    if (mode != 2) {
      hipMemsetAsync(stats, 0, (size_t)2 * O * sizeof(float), stream);
      vn_gemm_stats<<<grid, 128, 0, stream>>>(X, Wf, stats, O, Cin, P, xBS);
      bn_finalize<<<(O + 63) / 64, 64, 0, stream>>>(stats, O,
                                                    1.f / ((float)Bsz * (float)P));
    }
    if (mode == 0)
      vn_gemm_apply<0><<<grid, 128, 0, stream>>>(X, Wf, Wd, stats, out, O, Cin, P,
                                                 xBS, outBS);
    else if (mode == 1)
      vn_gemm_apply<1><<<grid, 128, 0, stream>>>(X, Wf, Wd, stats, out, O, Cin, P,
                                                 xBS, outBS);
    else
      vn_gemm_apply<2><<<grid, 128, 0, stream>>>(X, Wf, Wd, stats, out, O, Cin, P,
                                                 xBS, outBS);
  };

  // 1) KNN + edge features
  knn_kernel<<<dim3(Npts / 256, Bsz), 256, 0, stream>>>(x, IDX);
  const long totF = (long)Bsz * Npts * KNN;
  feat_kernel<<<(unsigned)((totF + 255) / 256), 256, 0, stream>>>(x, IDX, R3);

  // 2) pos VN layer over (N,k), then max-pool over k
  vn_layer(R3, pos_Wf, pos_Wd, R0, 21, 3, PK, 9LL * PK, 21LL * 3 * PK, 0);
  pool_k_kernel<<<dim3(Npts, Bsz), 256, 0, stream>>>(R0, pool_Wd, h21);

  // 3) c1
  vn_layer(h21, c1_Wf, c1_Wd, h21b, 21, 21, Npts, 21LL * 3 * Npts, 21LL * 3 * Npts, 0);

  // 4) STN feature-transform branch
  vn_layer(h21b, st1_Wf, st1_Wd, sA, 21, 21, Npts, 21LL * 3 * Npts, 21LL * 3 * Npts, 0);
  vn_layer(sA, st2_Wf, st2_Wd, sB, 42, 21, Npts, 21LL * 3 * Npts, 42LL * 3 * Npts, 0);
  vn_layer(sB, st3_Wf, st3_Wd, R1, 341, 42, Npts, 42LL * 3 * Npts, 341LL * 3 * Npts, 0);
  vn_layer(R1, stp_Wd, stp_Wd, R2, 341, 341, Npts, 341LL * 3 * Npts, 341LL * 3 * Npts, 2);
  pool_n_kernel<<<dim3(341, Bsz), 256, 0, stream>>>(R1, R2, sv341, 341);
  vn_small_kernel<<<1, 512, 0, stream>>>(sv341, stf1_Wf, stf1_Wd, sv170, 170, 341, 0);
  vn_small_kernel<<<1, 512, 0, stream>>>(sv170, stf2_Wf, stf2_Wd, sv85, 85, 170, 0);
  vn_small_kernel<<<1, 512, 0, stream>>>(sv85, stf3_W, stf3_W, sv21, 21, 85, 2);

  // 5) concat + c2 + c3(bn only, written into lower half of h682 region)
  const long tot42 = (long)Bsz * 42 * 3 * Npts;
  concat42_kernel<<<(unsigned)((tot42 + 255) / 256), 256, 0, stream>>>(h21b, sv21, hcat);
  vn_layer(hcat, c2_Wf, c2_Wd, h42, 42, 42, Npts, 42LL * 3 * Npts, 42LL * 3 * Npts, 0);
  vn_layer(h42, c3_W, c3_W, R0, 341, 42, Npts, 42LL * 3 * Npts, 682LL * 3 * Npts, 1);
  mean_bcast_kernel<<<dim3(341, Bsz), 256, 0, stream>>>(R0);  // rows 341..681 = mean

  // 6) VNStdFeature
  vn_layer(R0, std1_Wf, std1_Wd, R2, 341, 682, Npts, 682LL * 3 * Npts, 341LL * 3 * Npts, 0);
  vn_layer(R2, std2_Wf, std2_Wd, R3, 170, 341, Npts, 341LL * 3 * Npts, 170LL * 3 * Npts, 0);
  vn_layer(R3, std_lin, std_lin, z0b, 3, 170, Npts, 170LL * 3 * Npts, 9LL * Npts, 2);

  // 7) h_std = h . z0, max over N; outputs (out, z0) concatenated
  hstd_max_kernel<<<dim3(682, Bsz), 256, 0, stream>>>(R0, z0b, (float*)d_out);
  const long nz = (long)Bsz * 9 * Npts;
  copy_kernel<<<(unsigned)((nz + 255) / 256), 256, 0, stream>>>(
      z0b, (float*)d_out + (size_t)Bsz * 2046, nz);
}